// Involution_15642270892167
// MI455X (gfx1250) — compile-verified
//
#include <hip/hip_runtime.h>

// ---------------------------------------------------------------------------
// Fused involution for MI455X (gfx1250): wave32, WMMA f32 16x16x4, and the
// CDNA5 async global->LDS path (ASYNCcnt) for zero-padded x patch staging.
//   x:(8,256,56,56) -> [1x1 conv 256->64] -> BN -> ReLU -> [1x1 conv 64->784]
//   -> per-pixel 7x7 grouped (G=16) weighted gather -> out:(8,256,56,56)
// One block = one batch image x 64 pixels; wgt tile (784x64 f32) + t tile +
// double-buffered x patches all live in the 320KB CDNA5 LDS (~280KB used).
// ---------------------------------------------------------------------------

typedef float v2f __attribute__((ext_vector_type(2)));
typedef float v8f __attribute__((ext_vector_type(8)));

#define Cc   256          // input channels
#define CR   64           // reduced channels
#define CO   784          // 49*16 dynamic-weight channels
#define HH   56
#define WW   56
#define HWp  3136         // 56*56
#define PIX  64           // pixels per block (3136 = 49*64 exactly)
#define TSTR 72           // t LDS row stride (bank-conflict-free frag reads)
#define WSTR 64           // wgt LDS row stride
#define EPS  1e-5f

// Involution patch staging: 16 channels/chunk == one group; a 64-pixel tile
// spans exactly 2 image rows, so the 7x7 halo needs 8 rows x 62 cols.
#define PCH    16
#define PROWS  8
#define PCOLS  62
#define PATCHF (PCH * PROWS * PCOLS)      // 7936 floats per buffer

// LDS layout (floats)
#define OFF_WGT   0                          // 784*64 = 50176
#define OFF_T     (OFF_WGT + CO * WSTR)      // 64*72  = 4608
#define OFF_A     (OFF_T + CR * TSTR)        // 64
#define OFF_D     (OFF_A + CR)               // 64
#define OFF_B2    (OFF_D + CR)               // 784
#define OFF_PATCH (OFF_B2 + CO)              // 2*7936 = 15872
#define LDS_FLOATS (OFF_PATCH + 2 * PATCHF)  // 71568
#define LDS_BYTES  (LDS_FLOATS * 4)          // 286272 B  (< 320KB WGP LDS)

// ---- CDNA5 async global->LDS (GLOBAL_LOAD_ASYNC_TO_LDS_B32, ASYNCcnt) ----
__device__ __forceinline__ void async_ld_b32(unsigned ldsByteOff,
                                             const float* gaddr) {
    asm volatile("global_load_async_to_lds_b32 %0, %1, off"
                 :: "v"(ldsByteOff), "v"(gaddr) : "memory");
}
__device__ __forceinline__ void wait_async0() {
    asm volatile("s_wait_asynccnt 0" ::: "memory");
}

// Kick off the async DMA of one 16-channel zero-padded patch chunk.
// 7936 elements / 256 threads = 31 per thread; contiguous in col -> coalesced.
__device__ __forceinline__ void issue_patch_load(const float* __restrict__ xb,
                                                 int chunk, int r0row,
                                                 unsigned bufByteBase, int tid) {
    for (int q = 0; q < PATCHF / 256; ++q) {          // 31 iterations
        int idx  = tid + 256 * q;
        int ch_l = idx / (PROWS * PCOLS);
        int rem  = idx - ch_l * (PROWS * PCOLS);
        int r    = rem / PCOLS;
        int col  = rem - r * PCOLS;
        int irow = r0row - 3 + r;
        int icol = col - 3;
        if ((unsigned)irow < (unsigned)HH && (unsigned)icol < (unsigned)WW) {
            async_ld_b32(bufByteBase + (unsigned)idx * 4u,
                         xb + (size_t)(chunk * PCH + ch_l) * HWp
                            + irow * WW + icol);
        }
        // out-of-bounds halo positions keep their pre-zeroed value
    }
}

__global__ __launch_bounds__(256, 1)
void involution_fused(const float* __restrict__ x,
                      const float* __restrict__ w1,
                      const float* __restrict__ b1,
                      const float* __restrict__ gamma,
                      const float* __restrict__ beta,
                      const float* __restrict__ bn_mean,
                      const float* __restrict__ bn_var,
                      const float* __restrict__ w2,
                      const float* __restrict__ b2,
                      float* __restrict__ out)
{
    extern __shared__ float smem[];
    float* s_t  = smem + OFF_T;
    float* s_a  = smem + OFF_A;
    float* s_d  = smem + OFF_D;
    float* s_b2 = smem + OFF_B2;

    const int tid  = threadIdx.x;
    const int lane = tid & 31;        // wave32
    const int wv   = tid >> 5;        // 0..7
    const int lm   = lane & 15;       // M/N index within 16-wide tile
    const int lh   = lane >> 4;       // lane half (K pair / M half select)

    const int b     = blockIdx.x / 49;
    const int pix0  = (blockIdx.x % 49) * PIX;
    const int r0row = pix0 / WW;

    const float* xb = x + (size_t)b * (Cc * HWp);
    const unsigned patchBase = (unsigned)(uintptr_t)(smem + OFF_PATCH);

    // ---- Phase 0: fold BN into affine; stage b2; zero patch halo buffers --
    for (int i = tid; i < CR; i += 256) {
        float rs = rsqrtf(bn_var[i] + EPS);
        float aa = gamma[i] * rs;
        s_a[i] = aa;
        s_d[i] = (b1[i] - bn_mean[i]) * aa + beta[i];
    }
    for (int i = tid; i < CO; i += 256) s_b2[i] = b2[i];
    for (int i = tid; i < 2 * PATCHF; i += 256) smem[OFF_PATCH + i] = 0.0f;
    __syncthreads();

    // ---- Phase 1: GEMM1 (t = w1 . x_tile) via WMMA f32 16x16x4 -----------
    // 16 tiles (4M x 4N); wave wv: N-tile (wv&3), M-tiles (wv>>2), (wv>>2)+2,
    // sharing the B fragment between both M-tiles.
    {
        const int n1 = wv & 3;
        const int m1 = wv >> 2;
        const float* xt = xb + pix0;

        const float* aPtr0 = w1 + (size_t)(m1 * 16 + lm) * Cc + 2 * lh;
        const float* aPtr1 = w1 + (size_t)((m1 + 2) * 16 + lm) * Cc + 2 * lh;
        const float* bPtr  = xt + (size_t)(2 * lh) * HWp + n1 * 16 + lm;

        v8f acc0 = {}; v8f acc1 = {};
        for (int k0 = 0; k0 < Cc; k0 += 4) {
            v2f av0 = *(const v2f*)(aPtr0 + k0);
            v2f av1 = *(const v2f*)(aPtr1 + k0);
            v2f bv;
            bv.x = bPtr[(size_t)k0 * HWp];
            bv.y = bPtr[(size_t)(k0 + 1) * HWp];
            acc0 = __builtin_amdgcn_wmma_f32_16x16x4_f32(
                       false, av0, false, bv, (short)0, acc0, false, false);
            acc1 = __builtin_amdgcn_wmma_f32_16x16x4_f32(
                       false, av1, false, bv, (short)0, acc1, false, false);
        }
        #pragma unroll
        for (int v = 0; v < 8; ++v) {
            int r0 = m1 * 16 + lh * 8 + v;
            int r1 = (m1 + 2) * 16 + lh * 8 + v;
            float t0 = fmaxf(s_a[r0] * acc0[v] + s_d[r0], 0.0f);
            float t1 = fmaxf(s_a[r1] * acc1[v] + s_d[r1], 0.0f);
            s_t[r0 * TSTR + n1 * 16 + lm] = t0;
            s_t[r1 * TSTR + n1 * 16 + lm] = t1;
        }
    }
    __syncthreads();

    // Kick chunk-0 patch DMA now: overlaps with GEMM2 (disjoint LDS regions).
    issue_patch_load(xb, 0, r0row, patchBase, tid);

    // ---- Phase 2: GEMM2 (wgt = w2 . t), 196 tiles (49M x 4N) -------------
    // Wave owns N-tile (wv&3): hoist all 16 B fragments, loop M-tiles.
    {
        const int n2 = wv & 3;
        v2f bf[16];
        #pragma unroll
        for (int kk = 0; kk < 16; ++kk) {
            int kr = kk * 4 + 2 * lh;
            bf[kk].x = s_t[kr * TSTR + n2 * 16 + lm];
            bf[kk].y = s_t[(kr + 1) * TSTR + n2 * 16 + lm];
        }
        for (int m2 = (wv >> 2); m2 < 49; m2 += 2) {
            const float* aPtr = w2 + (size_t)(m2 * 16 + lm) * CR + 2 * lh;
            v8f cacc = {};
            #pragma unroll
            for (int kk = 0; kk < 16; ++kk) {
                v2f av = *(const v2f*)(aPtr + kk * 4);
                cacc = __builtin_amdgcn_wmma_f32_16x16x4_f32(
                           false, av, false, bf[kk], (short)0, cacc, false, false);
            }
            #pragma unroll
            for (int v = 0; v < 8; ++v) {
                int r = m2 * 16 + lh * 8 + v;
                smem[OFF_WGT + r * WSTR + n2 * 16 + lm] = cacc[v] + s_b2[r];
            }
        }
    }
    __syncthreads();

    // ---- Phase 3: involution gather, double-buffered LDS patches ---------
    // thread -> pixel (tid&63) + 4 channels {s, s+4, s+8, s+12} of the
    // chunk's group; wgt row is shared by all 16 channels of the group.
    {
        const int lp  = tid & 63;
        const int s   = tid >> 6;
        const int p   = pix0 + lp;
        const int ph  = p / WW;
        const int pw  = p - ph * WW;
        const int dph = ph - r0row;                 // 0 or 1
        float* ob = out + (size_t)b * (Cc * HWp);

        for (int c = 0; c < 16; ++c) {
            wait_async0();                          // chunk c DMA complete
            __syncthreads();                        // visible to all waves
            if (c < 15)                             // overlap next chunk DMA
                issue_patch_load(xb, c + 1, r0row,
                                 patchBase + (unsigned)(((c + 1) & 1) * PATCHF * 4),
                                 tid);

            const float* pb  = smem + OFF_PATCH + (c & 1) * PATCHF
                                    + s * (PROWS * PCOLS);
            const float* wgb = smem + OFF_WGT + (c * 49) * WSTR + lp;

            float a0 = 0.f, a1 = 0.f, a2 = 0.f, a3 = 0.f;
            #pragma unroll
            for (int i = 0; i < 7; ++i) {
                const float* prp = pb + (dph + i) * PCOLS + pw;
                #pragma unroll
                for (int j = 0; j < 7; ++j) {
                    float wvv = wgb[(i * 7 + j) * WSTR];
                    a0 = fmaf(wvv, prp[j               ], a0);
                    a1 = fmaf(wvv, prp[j +  4 * PROWS * PCOLS], a1);
                    a2 = fmaf(wvv, prp[j +  8 * PROWS * PCOLS], a2);
                    a3 = fmaf(wvv, prp[j + 12 * PROWS * PCOLS], a3);
                }
            }
            ob[(size_t)(c * PCH + s     ) * HWp + p] = a0;
            ob[(size_t)(c * PCH + s + 4 ) * HWp + p] = a1;
            ob[(size_t)(c * PCH + s + 8 ) * HWp + p] = a2;
            ob[(size_t)(c * PCH + s + 12) * HWp + p] = a3;
        }
    }
}

extern "C" void kernel_launch(void* const* d_in, const int* in_sizes, int n_in,
                              void* d_out, int out_size, void* d_ws, size_t ws_size,
                              hipStream_t stream) {
    const float* x       = (const float*)d_in[0];
    const float* w1      = (const float*)d_in[1];
    const float* b1      = (const float*)d_in[2];
    const float* gamma   = (const float*)d_in[3];
    const float* beta    = (const float*)d_in[4];
    const float* bn_mean = (const float*)d_in[5];
    const float* bn_var  = (const float*)d_in[6];
    const float* w2      = (const float*)d_in[7];
    const float* b2      = (const float*)d_in[8];
    float* out = (float*)d_out;

    // >64KB dynamic LDS (CDNA5 WGP has 320KB). Non-stream host call: safe
    // under graph capture, idempotent across calls.
    (void)hipFuncSetAttribute((const void*)involution_fused,
                              hipFuncAttributeMaxDynamicSharedMemorySize,
                              LDS_BYTES);

    involution_fused<<<dim3(8 * 49), dim3(256), LDS_BYTES, stream>>>(
        x, w1, b1, gamma, beta, bn_mean, bn_var, w2, b2, out);
}